// NCCLoss_61718680044281
// MI455X (gfx1250) — compile-verified
//
#include <hip/hip_runtime.h>

typedef __attribute__((ext_vector_type(2))) float v2f;
typedef __attribute__((ext_vector_type(4))) float v4f;
typedef __attribute__((ext_vector_type(8))) float v8f;

#define PITCH 28      // input LDS row pitch: 16B multiple (b128 stores), 28L%64 distinct L=0..15
#define VPITCH 20     // sV column pitch (col-major): 80B cols, 16B aligned, 20L%64 distinct
#define WAVES 4       // waves (tiles) per block
#define NPIX  8388608.0  // 32*512*512

__global__ void ncc_init(double* acc) {
    if (threadIdx.x == 0 && blockIdx.x == 0) *acc = 0.0;
}

__global__ void ncc_fin(const double* __restrict__ acc, float* __restrict__ out) {
    if (threadIdx.x == 0 && blockIdx.x == 0)
        out[0] = 1.0f - (float)(*acc / NPIX);
}

// one async b128 load: copies 16 bytes from global address %1 + off to LDS address %0 + off, tracked by ASYNCcnt
#define ASYNC_LDS_B128(lds, ga, off)                                           \
    asm volatile("global_load_async_to_lds_b128 %0, %1, off offset:" #off      \
                 :: "v"(lds), "v"(ga) : "memory")

__global__ __launch_bounds__(32 * WAVES) void ncc_main(const float* __restrict__ I,
                                                       const float* __restrict__ J,
                                                       double* __restrict__ acc)
{
    // Per-wave LDS: 24x24 halo inputs (row-major), 5x 24x16 vertical sums (col-major)
    __shared__ __align__(16) float sI[WAVES][24 * PITCH];
    __shared__ __align__(16) float sJ[WAVES][24 * PITCH];
    __shared__ __align__(16) float sV[WAVES][5][24 * VPITCH];

    const int lane = threadIdx.x & 31;
    const int w    = threadIdx.x >> 5;

    // tile id -> (batch, tile-y, tile-x); 32x32 tiles per 512x512 image, 32 images
    const int tile = blockIdx.x * WAVES + w;
    const int tx = tile & 31;
    const int ty = (tile >> 5) & 31;
    const int b  = tile >> 10;
    const int x0 = tx * 16 - 4;     // halo origin (can be negative)
    const int y0 = ty * 16 - 4;
    const size_t base = (size_t)b * (512 * 512);

    float* smI = sI[w];
    float* smJ = sJ[w];

    const bool inner = (tx >= 1) && (tx <= 30) && (ty >= 1) && (ty <= 30);
    if (inner) {
        // ---- Fast path: 88% of tiles. Lanes 0-23 each stage one 24-float halo row
        // for I and J via async global->LDS b128 copies (no VGPR round trip).
        if (lane < 24) {
            const size_t g = base + (size_t)(y0 + lane) * 512 + x0;  // 16B aligned
            unsigned ldsI = (unsigned)(uintptr_t)&smI[lane * PITCH];
            unsigned ldsJ = (unsigned)(uintptr_t)&smJ[lane * PITCH];
            unsigned long long gaI = (unsigned long long)(uintptr_t)(I + g);
            unsigned long long gaJ = (unsigned long long)(uintptr_t)(J + g);
            ASYNC_LDS_B128(ldsI, gaI, 0);
            ASYNC_LDS_B128(ldsI, gaI, 16);
            ASYNC_LDS_B128(ldsI, gaI, 32);
            ASYNC_LDS_B128(ldsI, gaI, 48);
            ASYNC_LDS_B128(ldsI, gaI, 64);
            ASYNC_LDS_B128(ldsI, gaI, 80);
            ASYNC_LDS_B128(ldsJ, gaJ, 0);
            ASYNC_LDS_B128(ldsJ, gaJ, 16);
            ASYNC_LDS_B128(ldsJ, gaJ, 32);
            ASYNC_LDS_B128(ldsJ, gaJ, 48);
            ASYNC_LDS_B128(ldsJ, gaJ, 64);
            ASYNC_LDS_B128(ldsJ, gaJ, 80);
        }
    } else {
        // ---- Edge path: zero-padded, bounds-checked scalar staging
        #pragma unroll
        for (int it = 0; it < 18; ++it) {
            int idx = it * 32 + lane;
            int r = idx / 24;
            int c = idx - r * 24;
            int gy = y0 + r, gx = x0 + c;
            bool ok = (gy >= 0) && (gy < 512) && (gx >= 0) && (gx < 512);
            size_t g = base + (size_t)gy * 512 + gx;
            float vi = ok ? I[g] : 0.0f;
            float vj = ok ? J[g] : 0.0f;
            smI[r * PITCH + c] = vi;
            smJ[r * PITCH + c] = vj;
        }
    }
    asm volatile("s_wait_asynccnt 0" ::: "memory");  // async LDS writes visible
    __builtin_amdgcn_wave_barrier();

    const int m     = lane & 15;          // M (or N) index within 16
    const int khalf = (lane >> 4) << 1;   // K sub-offset: 0 for lanes 0-15, 2 for 16-31
    const int row0  = (lane >> 4) << 3;   // C/D row base: 0 or 8

    // ---- Banded-ones operand, identical for pass V (as A) and pass H (as B):
    // band[i][m][kk] = (m <= kk <= m+8), kk = 4*i + khalf (+1 for .y)
    v2f band[6];
    #pragma unroll
    for (int i = 0; i < 6; ++i) {
        const int kk = i * 4 + khalf;
        band[i].x = (kk     >= m && kk     <= m + 8) ? 1.0f : 0.0f;
        band[i].y = (kk + 1 >= m && kk + 1 <= m + 8) ? 1.0f : 0.0f;
    }

    // ---- Pass V: vertical 9-tap box sum as  out_v[16x24] = BandV[16x24] x region[24x24]
    // Two 16-column chunks (cb = 0, 8) cover the 24 halo columns.
    #pragma unroll
    for (int chunk = 0; chunk < 2; ++chunk) {
        const int cb  = chunk * 8;
        const int col = cb + m;
        v8f aI = {}, aJ = {}, aI2 = {}, aJ2 = {}, aIJ = {};
        #pragma unroll
        for (int i = 0; i < 6; ++i) {
            const int kk = i * 4 + khalf;
            v2f bI, bJ;
            bI.x = smI[kk * PITCH + col];
            bI.y = smI[(kk + 1) * PITCH + col];
            bJ.x = smJ[kk * PITCH + col];
            bJ.y = smJ[(kk + 1) * PITCH + col];
            v2f bI2 = bI * bI;
            v2f bJ2 = bJ * bJ;
            v2f bIJ = bI * bJ;
            aI  = __builtin_amdgcn_wmma_f32_16x16x4_f32(false, band[i], false, bI,  (short)0, aI,  false, false);
            aJ  = __builtin_amdgcn_wmma_f32_16x16x4_f32(false, band[i], false, bJ,  (short)0, aJ,  false, false);
            aI2 = __builtin_amdgcn_wmma_f32_16x16x4_f32(false, band[i], false, bI2, (short)0, aI2, false, false);
            aJ2 = __builtin_amdgcn_wmma_f32_16x16x4_f32(false, band[i], false, bJ2, (short)0, aJ2, false, false);
            aIJ = __builtin_amdgcn_wmma_f32_16x16x4_f32(false, band[i], false, bIJ, (short)0, aIJ, false, false);
        }
        // Col-major sV: a lane's 8 D values (rows row0..row0+7 at fixed col) are
        // contiguous -> two ds_store_b128 per quantity.
        const int o = col * VPITCH + row0;   // 16B aligned
        #pragma unroll
        for (int q = 0; q < 5; ++q) {
            const v8f* accq[5] = { &aI, &aJ, &aI2, &aJ2, &aIJ };
            const v8f a = *accq[q];
            *(v4f*)&sV[w][q][o]     = __builtin_shufflevector(a, a, 0, 1, 2, 3);
            *(v4f*)&sV[w][q][o + 4] = __builtin_shufflevector(a, a, 4, 5, 6, 7);
        }
    }
    __builtin_amdgcn_wave_barrier();

    // ---- Pass H: horizontal 9-tap box sum as  out[16x16] = out_v[16x24] x BandH[24x16]
    v8f fI = {}, fJ = {}, fI2 = {}, fJ2 = {}, fIJ = {};
    #pragma unroll
    for (int i = 0; i < 6; ++i) {
        const int kk  = i * 4 + khalf;
        const int rb0 = kk * VPITCH + m;        // col kk, row m (col-major)
        const int rb1 = rb0 + VPITCH;           // col kk+1 -> ds_load_2addr pair
        v2f aIv, aJv, aI2v, aJ2v, aIJv;
        aIv.x  = sV[w][0][rb0];  aIv.y  = sV[w][0][rb1];
        aJv.x  = sV[w][1][rb0];  aJv.y  = sV[w][1][rb1];
        aI2v.x = sV[w][2][rb0];  aI2v.y = sV[w][2][rb1];
        aJ2v.x = sV[w][3][rb0];  aJ2v.y = sV[w][3][rb1];
        aIJv.x = sV[w][4][rb0];  aIJv.y = sV[w][4][rb1];
        fI  = __builtin_amdgcn_wmma_f32_16x16x4_f32(false, aIv,  false, band[i], (short)0, fI,  false, false);
        fJ  = __builtin_amdgcn_wmma_f32_16x16x4_f32(false, aJv,  false, band[i], (short)0, fJ,  false, false);
        fI2 = __builtin_amdgcn_wmma_f32_16x16x4_f32(false, aI2v, false, band[i], (short)0, fI2, false, false);
        fJ2 = __builtin_amdgcn_wmma_f32_16x16x4_f32(false, aJ2v, false, band[i], (short)0, fJ2, false, false);
        fIJ = __builtin_amdgcn_wmma_f32_16x16x4_f32(false, aIJv, false, band[i], (short)0, fIJ, false, false);
    }

    // ---- Per-pixel NCC (simplified exactly: cross = IJ - Is*Js/W, var = S2 - S^2/W)
    const float invW = 1.0f / 81.0f;
    float local = 0.0f;
    #pragma unroll
    for (int v = 0; v < 8; ++v) {
        float Is  = fI[v],  Js  = fJ[v];
        float I2s = fI2[v], J2s = fJ2[v], IJs = fIJ[v];
        float cross = IJs - Is * Js * invW;
        float Iv    = I2s - Is * Is * invW;
        float Jv    = J2s - Js * Js * invW;
        local += cross * cross / (Iv * Jv + 1e-6f);
    }

    // ---- Wave reduction + one double atomic per wave
    #pragma unroll
    for (int off = 16; off > 0; off >>= 1)
        local += __shfl_xor(local, off, 32);
    if (lane == 0)
        atomicAdd(acc, (double)local);
}

extern "C" void kernel_launch(void* const* d_in, const int* in_sizes, int n_in,
                              void* d_out, int out_size, void* d_ws, size_t ws_size,
                              hipStream_t stream) {
    const float* I = (const float*)d_in[0];   // predict
    const float* J = (const float*)d_in[1];   // target
    double* acc = (double*)d_ws;
    float*  out = (float*)d_out;

    hipLaunchKernelGGL(ncc_init, dim3(1), dim3(32), 0, stream, acc);
    // 32768 tiles (32 imgs x 32x32 tiles), 4 tiles per 128-thread block
    hipLaunchKernelGGL(ncc_main, dim3(8192), dim3(32 * WAVES), 0, stream, I, J, acc);
    hipLaunchKernelGGL(ncc_fin, dim3(1), dim3(32), 0, stream, acc, out);
}